// CrossLayer_67551245631680
// MI455X (gfx1250) — compile-verified
//
#include <hip/hip_runtime.h>

typedef __attribute__((ext_vector_type(2))) float v2f;
typedef __attribute__((ext_vector_type(4))) float f32x4;
typedef __attribute__((ext_vector_type(8))) float v8f;

#define HAS_WMMA_F32X4 __has_builtin(__builtin_amdgcn_wmma_f32_16x16x4_f32)

// ---------------------------------------------------------------------------
// Phase 1: s[row] = dot(xl[row, :], W)  via V_WMMA_F32_16X16X4_F32
//   Block = 256 threads = 8 waves; each wave owns 16 rows.
//   A (16x4 f32): lane l -> row m0 + (l&15), K = 2*(l>>4) + {0,1}  (float2)
//   B (4x16 f32): broadcast W chunk, same K mapping per lane half  (float2)
//   D (16x16 f32): every column equal; lane 0 holds M=0..7 in acc[0..7],
//                  lane 16 holds M=8..15.
// ---------------------------------------------------------------------------
__global__ __launch_bounds__(256) void cross_rowdot_wmma(
    const float* __restrict__ xl, const float* __restrict__ W,
    float* __restrict__ s, int d)
{
    __shared__ float Wlds[4096];
    const int tid = threadIdx.x;
    for (int i = tid; i < d; i += 256) Wlds[i] = W[i];
    __syncthreads();

    const int wave = tid >> 5;
    const int lane = tid & 31;
    const int m0   = blockIdx.x * 128 + wave * 16;
    const int kOff = (lane >> 4) << 1;                 // 0 or 2
    const float* xrow = xl + (size_t)(m0 + (lane & 15)) * (size_t)d + kOff;
    const float* wp   = Wlds + kOff;

#if HAS_WMMA_F32X4
    v8f acc0 = {};
    v8f acc1 = {};
    for (int k0 = 0; k0 < d; k0 += 8) {
        v2f a0 = *(const v2f*)(xrow + k0);
        v2f b0 = *(const v2f*)(wp   + k0);
        v2f a1 = *(const v2f*)(xrow + k0 + 4);
        v2f b1 = *(const v2f*)(wp   + k0 + 4);
        acc0 = __builtin_amdgcn_wmma_f32_16x16x4_f32(
            false, a0, false, b0, (short)0, acc0, false, false);
        acc1 = __builtin_amdgcn_wmma_f32_16x16x4_f32(
            false, a1, false, b1, (short)0, acc1, false, false);
    }
    v8f acc = acc0 + acc1;
    if ((lane & 15) == 0) {
        const int base = m0 + ((lane >> 4) << 3);      // lane0 -> m0, lane16 -> m0+8
        #pragma unroll
        for (int r = 0; r < 8; ++r) s[base + r] = acc[r];
    }
#else
    // Fallback: per-lane partial (same data assignment), combine lane l with l+16.
    float p = 0.0f;
    for (int k0 = 0; k0 < d; k0 += 4)
        p += xrow[k0] * wp[k0] + xrow[k0 + 1] * wp[k0 + 1];
    p += __shfl_xor(p, 16, 32);
    if (lane < 16) s[m0 + lane] = p;
#endif
}

// ---------------------------------------------------------------------------
// Phase 2: out[b][j] = x0[b][j] * s[b] + bias[j] + xl[b][j]
//   Fully coalesced float4 streaming; NT hints on the 768 MB streams
//   (exceeds 192 MB L2), RT on bias (16 KB, reused by all rows).
//   d = 4096 -> 1024 float4 per row.
// ---------------------------------------------------------------------------
__global__ __launch_bounds__(256) void cross_fuse_stream(
    const float* __restrict__ x0, const float* __restrict__ xl,
    const float* __restrict__ bias, const float* __restrict__ s,
    float* __restrict__ out)
{
    const size_t i   = (size_t)blockIdx.x * 256 + threadIdx.x; // float4 index
    const int    row = (int)(i >> 10);
    const int    c4  = (int)(i & 1023);
    const size_t off = i << 2;

    const float sv = s[row];
    f32x4 a  = __builtin_nontemporal_load((const f32x4*)(x0 + off));
    f32x4 l  = __builtin_nontemporal_load((const f32x4*)(xl + off));
    f32x4 bb = *(const f32x4*)(bias + ((size_t)c4 << 2));

    f32x4 o = a * sv + bb + l;
    __builtin_nontemporal_store(o, (f32x4*)(out + off));
}

extern "C" void kernel_launch(void* const* d_in, const int* in_sizes, int n_in,
                              void* d_out, int out_size, void* d_ws, size_t ws_size,
                              hipStream_t stream) {
    const float* x0   = (const float*)d_in[0];
    const float* xl   = (const float*)d_in[1];
    const float* W    = (const float*)d_in[2];
    const float* bias = (const float*)d_in[3];
    float*       out  = (float*)d_out;
    float*       s    = (float*)d_ws;              // B floats of scratch

    const int d = in_sizes[2];                      // 4096
    const int B = in_sizes[0] / d;                  // 16384

    // Phase 1: 8 waves/block * 16 rows/wave = 128 rows per block.
    cross_rowdot_wmma<<<B / 128, 256, 0, stream>>>(xl, W, s, d);

    // Phase 2: one float4 per thread.
    const unsigned nvec4 = (unsigned)(((size_t)B * (size_t)d) >> 2);
    cross_fuse_stream<<<nvec4 / 256, 256, 0, stream>>>(x0, xl, bias, s, out);
}